// LinearAttention_47871705481538
// MI455X (gfx1250) — compile-verified
//
#include <hip/hip_runtime.h>
#include <math.h>

typedef __attribute__((ext_vector_type(2))) float v2f;
typedef __attribute__((ext_vector_type(4))) float v4f;
typedef __attribute__((ext_vector_type(8))) float v8f;

#define NPIX  65536
#define DIMC  128
#define QKVR  384
#define HEADS 4
#define DHEAD 32
#define LDSS  68   // LDS row stride (floats), avoids bank conflicts
#define ATTN_SCALE 0.17677669529663687f   // 32^-0.5
#define NORMS 11.313708498984760f         // 128^0.5

// D = A(16x4) * B(4x16) + C, all f32, exact-precision WMMA
__device__ __forceinline__ v8f wmma4(v2f a, v2f b, v8f c) {
  return __builtin_amdgcn_wmma_f32_16x16x4_f32(false, a, false, b, (short)0, c, false, false);
}

// ---------- prep: fold (gamma1+1) into w_qkv; zero ctx accumulator ----------
__global__ void k_prep(const float* __restrict__ w_qkv, const float* __restrict__ g1,
                       float* __restrict__ wfold, float* __restrict__ ctx) {
  int i = blockIdx.x * 256 + threadIdx.x;
  if (i < QKVR * DIMC) {
    int c = i & (DIMC - 1);
    wfold[i] = w_qkv[i] * (g1[c] + 1.0f);
  }
  if (i < HEADS * DHEAD * DHEAD) ctx[i] = 0.0f;
}

// ---------- per-pixel RMS inverse norm: inv[p] = sqrt(128)/||src[:,p]|| ----------
__global__ void k_rms_inv(const float* __restrict__ src, float* __restrict__ inv) {
  int p = blockIdx.x * 256 + threadIdx.x;
  float s = 0.0f;
  #pragma unroll 4
  for (int c = 0; c < DIMC; ++c) {
    float v = src[c * NPIX + p];
    s += v * v;
  }
  inv[p] = NORMS / sqrtf(s);
}

// ---------- qkv GEMM: qkv(384,N) = wfold(384,128) @ (x * inv1) ----------
// block = 256 threads (8 waves); block computes all 384 rows for a 64-pixel tile.
__global__ void __launch_bounds__(256) k_qkv(const float* __restrict__ x,
                                             const float* __restrict__ inv1,
                                             const float* __restrict__ wfold,
                                             float* __restrict__ qkv) {
  __shared__ float xs[DIMC * LDSS];
  const int t = threadIdx.x;
  const int nBase = blockIdx.x * 64;
  const int col4 = (t & 15) * 4;
  const int r0 = t >> 4;

  v4f iv = *(const v4f*)(inv1 + nBase + col4);
  #pragma unroll
  for (int i = 0; i < 8; ++i) {
    int r = r0 + i * 16;
    v4f xv = *(const v4f*)(x + r * NPIX + nBase + col4);
    xv *= iv;
    *(v4f*)(&xs[r * LDSS + col4]) = xv;
  }
  __syncthreads();

  const int wave = t >> 5;
  const int lane = t & 31;
  const int lm = lane & 15;
  const int lh = lane >> 4;

  v8f acc[3][4];
  #pragma unroll
  for (int m = 0; m < 3; ++m)
    #pragma unroll
    for (int n = 0; n < 4; ++n)
      acc[m][n] = (v8f){0,0,0,0,0,0,0,0};

  for (int kk = 0; kk < 32; ++kk) {
    const int k0 = kk * 4;
    v2f a[3];
    #pragma unroll
    for (int m = 0; m < 3; ++m) {
      int row = (wave * 3 + m) * 16 + lm;
      a[m] = *(const v2f*)(wfold + row * DIMC + k0 + lh * 2);
    }
    #pragma unroll
    for (int n = 0; n < 4; ++n) {
      v2f b;
      b.x = xs[(k0 + lh * 2) * LDSS + n * 16 + lm];
      b.y = xs[(k0 + lh * 2 + 1) * LDSS + n * 16 + lm];
      #pragma unroll
      for (int m = 0; m < 3; ++m)
        acc[m][n] = wmma4(a[m], b, acc[m][n]);
    }
  }

  #pragma unroll
  for (int m = 0; m < 3; ++m) {
    const int mBase = (wave * 3 + m) * 16;
    #pragma unroll
    for (int n = 0; n < 4; ++n)
      #pragma unroll
      for (int i = 0; i < 8; ++i)
        qkv[(mBase + i + lh * 8) * NPIX + nBase + n * 16 + lm] = acc[m][n][i];
  }
}

// ---------- k row max (incl. 4 memory tokens) ----------
__global__ void k_kmax(const float* __restrict__ kmat, const float* __restrict__ memk,
                       float* __restrict__ kmax) {
  __shared__ float red[256];
  const int row = blockIdx.x;           // h*32+d
  const int t = threadIdx.x;
  const float* kr = kmat + row * NPIX;
  float m = -1e30f;
  for (int i = t; i < NPIX; i += 256) m = fmaxf(m, kr[i]);
  if (t < 4) m = fmaxf(m, memk[row * 4 + t]);
  red[t] = m; __syncthreads();
  for (int s = 128; s > 0; s >>= 1) {
    if (t < s) red[t] = fmaxf(red[t], red[t + s]);
    __syncthreads();
  }
  if (t == 0) kmax[row] = red[0];
}

// ---------- k row sum of exp (incl. memory tokens) ----------
__global__ void k_ksum(const float* __restrict__ kmat, const float* __restrict__ memk,
                       const float* __restrict__ kmax, float* __restrict__ ksum) {
  __shared__ float red[256];
  const int row = blockIdx.x;
  const int t = threadIdx.x;
  const float km = kmax[row];
  const float* kr = kmat + row * NPIX;
  float s = 0.0f;
  for (int i = t; i < NPIX; i += 256) s += expf(kr[i] - km);
  if (t < 4) s += expf(memk[row * 4 + t] - km);
  red[t] = s; __syncthreads();
  for (int st = 128; st > 0; st >>= 1) {
    if (t < st) red[t] += red[t + st];
    __syncthreads();
  }
  if (t == 0) ksum[row] = red[0];
}

// ---------- context: ctx[h,d,e] += sum_n exp(k[h,d,n]-kmax) * v[h,e,n] ----------
// grid = (32 chunks, 4 heads), block = 128 threads (4 waves -> 2x2 16x16 tiles)
__global__ void __launch_bounds__(128) k_ctx(const float* __restrict__ kmat,
                                             const float* __restrict__ vmat,
                                             const float* __restrict__ kmax,
                                             float* __restrict__ ctx) {
  __shared__ float ks[DHEAD * LDSS];
  __shared__ float vs[DHEAD * LDSS];
  const int head = blockIdx.y;
  const int n0base = blockIdx.x * 2048;
  const int t = threadIdx.x;
  const int wave = t >> 5;
  const int lane = t & 31;
  const int lm = lane & 15;
  const int lh = lane >> 4;
  const int mTile = wave & 1;   // d tile
  const int eTile = wave >> 1;  // e tile
  const int col4 = (t & 15) * 4;
  const int r0 = t >> 4;        // 0..7

  v8f acc = (v8f){0,0,0,0,0,0,0,0};

  for (int sub = 0; sub < 32; ++sub) {
    const int n0 = n0base + sub * 64;
    #pragma unroll
    for (int i = 0; i < 4; ++i) {
      int r = r0 + i * 8;
      float km = kmax[head * DHEAD + r];
      v4f kv = *(const v4f*)(kmat + (head * DHEAD + r) * NPIX + n0 + col4);
      kv.x = expf(kv.x - km); kv.y = expf(kv.y - km);
      kv.z = expf(kv.z - km); kv.w = expf(kv.w - km);
      *(v4f*)(&ks[r * LDSS + col4]) = kv;
      *(v4f*)(&vs[r * LDSS + col4]) = *(const v4f*)(vmat + (head * DHEAD + r) * NPIX + n0 + col4);
    }
    __syncthreads();
    #pragma unroll
    for (int kk = 0; kk < 16; ++kk) {
      const int k0 = kk * 4;
      v2f a, b;
      a.x = ks[(mTile * 16 + lm) * LDSS + k0 + lh * 2];
      a.y = ks[(mTile * 16 + lm) * LDSS + k0 + lh * 2 + 1];
      b.x = vs[(eTile * 16 + lm) * LDSS + k0 + lh * 2];
      b.y = vs[(eTile * 16 + lm) * LDSS + k0 + lh * 2 + 1];
      acc = wmma4(a, b, acc);
    }
    __syncthreads();
  }
  #pragma unroll
  for (int i = 0; i < 8; ++i)
    atomicAdd(&ctx[(head * DHEAD + mTile * 16 + i + lh * 8) * DHEAD + eTile * 16 + lm], acc[i]);
}

// ---------- memory-token contribution to ctx ----------
__global__ void k_ctx_mem(const float* __restrict__ memk, const float* __restrict__ memv,
                          const float* __restrict__ kmax, float* __restrict__ ctx) {
  int idx = blockIdx.x * 256 + threadIdx.x;
  if (idx < HEADS * DHEAD * DHEAD) {
    int e = idx & 31;
    int hd = idx >> 5;       // h*32+d
    int h = hd >> 5;
    float km = kmax[hd];
    float s = 0.0f;
    #pragma unroll
    for (int m = 0; m < 4; ++m)
      s += expf(memk[hd * 4 + m] - km) * memv[(h * DHEAD + e) * 4 + m];
    ctx[hd * DHEAD + e] += s;
  }
}

// ---------- W2[o,hd] = SCALE/ksum[hd] * sum_e w_out[o,h*32+e]*ctx[hd*32+e] ----------
__global__ void k_w2(const float* __restrict__ w_out, const float* __restrict__ ctx,
                     const float* __restrict__ ksum, float* __restrict__ W2) {
  int i = blockIdx.x * 256 + threadIdx.x;
  if (i < DIMC * DIMC) {
    int o = i >> 7;
    int hd = i & 127;
    int h = hd >> 5;
    float s = 0.0f;
    #pragma unroll 8
    for (int e = 0; e < DHEAD; ++e)
      s += w_out[o * DIMC + h * DHEAD + e] * ctx[hd * DHEAD + e];
    W2[o * DIMC + hd] = s * ATTN_SCALE / ksum[hd];
  }
}

// ---------- fused output GEMM: out(128,N) = W2 @ softmax_d(q) + b ----------
__global__ void __launch_bounds__(256) k_out(const float* __restrict__ qmat,
                                             const float* __restrict__ W2,
                                             const float* __restrict__ b_out,
                                             float* __restrict__ out) {
  __shared__ float qs[DIMC * LDSS];
  const int t = threadIdx.x;
  const int nBase = blockIdx.x * 64;
  const int col4 = (t & 15) * 4;
  const int r0 = t >> 4;

  #pragma unroll
  for (int i = 0; i < 8; ++i) {
    int r = r0 + i * 16;
    *(v4f*)(&qs[r * LDSS + col4]) = *(const v4f*)(qmat + r * NPIX + nBase + col4);
  }
  __syncthreads();

  // softmax over the 32 feature rows within each head, per pixel column
  {
    const int h = t >> 6;        // 0..3
    const int p = t & 63;        // 0..63
    const int base = h * DHEAD * LDSS + p;
    float mx = -1e30f;
    #pragma unroll 8
    for (int d = 0; d < DHEAD; ++d) mx = fmaxf(mx, qs[base + d * LDSS]);
    float s = 0.0f;
    #pragma unroll 8
    for (int d = 0; d < DHEAD; ++d) {
      float e = expf(qs[base + d * LDSS] - mx);
      qs[base + d * LDSS] = e;
      s += e;
    }
    float invs = 1.0f / s;
    #pragma unroll 8
    for (int d = 0; d < DHEAD; ++d) qs[base + d * LDSS] *= invs;
  }
  __syncthreads();

  const int wave = t >> 5;
  const int lane = t & 31;
  const int lm = lane & 15;
  const int lh = lane >> 4;

  v8f acc[4];
  #pragma unroll
  for (int n = 0; n < 4; ++n) acc[n] = (v8f){0,0,0,0,0,0,0,0};

  for (int kk = 0; kk < 32; ++kk) {
    const int k0 = kk * 4;
    v2f a = *(const v2f*)(W2 + (wave * 16 + lm) * DIMC + k0 + lh * 2);
    #pragma unroll
    for (int n = 0; n < 4; ++n) {
      v2f b;
      b.x = qs[(k0 + lh * 2) * LDSS + n * 16 + lm];
      b.y = qs[(k0 + lh * 2 + 1) * LDSS + n * 16 + lm];
      acc[n] = wmma4(a, b, acc[n]);
    }
  }
  #pragma unroll
  for (int n = 0; n < 4; ++n)
    #pragma unroll
    for (int i = 0; i < 8; ++i) {
      int orow = wave * 16 + i + lh * 8;
      out[orow * NPIX + nBase + n * 16 + lm] = acc[n][i] + b_out[orow];
    }
}

// ---------- final in-place RMS scale: out *= inv2[p]*(gamma2[c]+1) ----------
__global__ void k_scale(float* __restrict__ out, const float* __restrict__ inv2,
                        const float* __restrict__ g2) {
  int i = blockIdx.x * 256 + threadIdx.x;
  int c = i >> 16;
  int p = i & (NPIX - 1);
  out[i] = out[i] * inv2[p] * (g2[c] + 1.0f);
}

// ---------------- host launcher ----------------
extern "C" void kernel_launch(void* const* d_in, const int* in_sizes, int n_in,
                              void* d_out, int out_size, void* d_ws, size_t ws_size,
                              hipStream_t stream) {
  (void)in_sizes; (void)n_in; (void)out_size; (void)ws_size;
  const float* x     = (const float*)d_in[0];
  const float* g1    = (const float*)d_in[1];
  const float* w_qkv = (const float*)d_in[2];
  const float* memkv = (const float*)d_in[3];   // (2,4,32,4)
  const float* w_out = (const float*)d_in[4];
  const float* b_out = (const float*)d_in[5];
  const float* g2    = (const float*)d_in[6];
  float* out = (float*)d_out;
  float* ws  = (float*)d_ws;

  // workspace layout (float offsets)
  float* inv1  = ws;                 // 65536
  float* inv2  = ws + 65536;         // 65536
  float* wfold = ws + 131072;        // 49152
  float* W2    = ws + 180224;        // 16384
  float* kmax  = ws + 196608;        // 128
  float* ksum  = ws + 196736;        // 128
  float* ctx   = ws + 196864;        // 4096
  float* qkv   = ws + 262144;        // 384*65536
  float* qmat  = qkv;
  float* kmat  = qkv + DIMC * NPIX;
  float* vmat  = qkv + 2 * DIMC * NPIX;
  const float* memk = memkv;         // (4,32,4)
  const float* memv = memkv + 512;

  k_prep   <<<192, 256, 0, stream>>>(w_qkv, g1, wfold, ctx);
  k_rms_inv<<<NPIX / 256, 256, 0, stream>>>(x, inv1);
  k_qkv    <<<NPIX / 64, 256, 0, stream>>>(x, inv1, wfold, qkv);
  k_kmax   <<<128, 256, 0, stream>>>(kmat, memk, kmax);
  k_ksum   <<<128, 256, 0, stream>>>(kmat, memk, kmax, ksum);
  k_ctx    <<<dim3(32, HEADS), 128, 0, stream>>>(kmat, vmat, kmax, ctx);
  k_ctx_mem<<<16, 256, 0, stream>>>(memk, memv, kmax, ctx);
  k_w2     <<<64, 256, 0, stream>>>(w_out, ctx, ksum, W2);
  k_out    <<<NPIX / 64, 256, 0, stream>>>(qmat, W2, b_out, out);
  k_rms_inv<<<NPIX / 256, 256, 0, stream>>>(out, inv2);
  k_scale  <<<DIMC * NPIX / 256, 256, 0, stream>>>(out, inv2, g2);
}